// MolTEmbeddings_50800873177193
// MI455X (gfx1250) — compile-verified
//
#include <hip/hip_runtime.h>
#include <hip/hip_bf16.h>

namespace {
constexpr int B   = 64;
constexpr int L   = 512;
constexpr int D   = 768;
constexpr int K   = 2;
constexpr int P   = 128;
constexpr int HID = D + K * P;   // 1024
constexpr int NTOK = B * L;      // 32768
constexpr int TPB  = 256;        // threads per block (8 wave32)
constexpr int TOKS = 8;          // tokens per block
constexpr float EPS = 1e-12f;
}

#if defined(__HIP_DEVICE_COMPILE__) && defined(__gfx1250__) && \
    __has_builtin(__builtin_amdgcn_global_load_async_to_lds_b128) && \
    __has_builtin(__builtin_amdgcn_s_wait_asynccnt)
#define MOLT_ASYNC_LDS 1
#else
#define MOLT_ASYNC_LDS 0
#endif

#if MOLT_ASYNC_LDS
// Builtin signature (from hipcc diagnostic): parameters are
// 'int __attribute__((vector_size(16)))' pointers in addrspace(1)/(3).
typedef int molt_v4i __attribute__((vector_size(16)));
typedef __attribute__((address_space(1))) molt_v4i molt_gv4i;
typedef __attribute__((address_space(3))) molt_v4i molt_lv4i;
#endif

__device__ __forceinline__ float block_sum(float x, float* red, int lane, int wid) {
  // wave32 reduction (gfx1250 is wave32-only)
  #pragma unroll
  for (int off = 16; off > 0; off >>= 1)
    x += __shfl_xor(x, off, 32);
  if (lane == 0) red[wid] = x;
  __syncthreads();
  float r = 0.f;
  #pragma unroll
  for (int w = 0; w < TPB / 32; ++w) r += red[w];
  __syncthreads();   // safe to reuse `red`
  return r;
}

__global__ __launch_bounds__(TPB) void molt_embed_ln_kernel(
    const int*   __restrict__ input_ids,        // [B,L]
    const int*   __restrict__ token_type_ids,   // [B,L]
    const int*   __restrict__ pos_embed_ids,    // [B,L,K]
    const float* __restrict__ lp_embeds,        // [B,L,P]
    const int*   __restrict__ p_ring,
    const int*   __restrict__ p_charge,
    const int*   __restrict__ p_hyb,
    const int*   __restrict__ p_chir,
    const int*   __restrict__ p_arom,
    const int*   __restrict__ p_conj,
    const int*   __restrict__ p_stereo,
    const float* __restrict__ mol_features,     // [B,L]
    const float* __restrict__ target_values,    // [B,L]
    const float* __restrict__ emb_table,        // [V,D]
    const float* __restrict__ type_table,       // [8,D]
    const float* __restrict__ ring_tab,         // [3,D]
    const float* __restrict__ charge_tab,       // [4,D]
    const float* __restrict__ hyb_tab,          // [9,D]
    const float* __restrict__ chir_tab,         // [5,D]
    const float* __restrict__ arom_tab,         // [3,D]
    const float* __restrict__ conj_tab,         // [3,D]
    const float* __restrict__ stereo_tab,       // [7,D]
    const float* __restrict__ ln_gamma,         // [HID]
    const float* __restrict__ ln_beta,          // [HID]
    float*       __restrict__ out)              // [B,L,HID]
{
  __shared__ __align__(16) float s_gamma[HID];
  __shared__ __align__(16) float s_beta[HID];
  __shared__ float s_red[TPB / 32];

  const int tid  = threadIdx.x;
  const int lane = tid & 31;
  const int wid  = tid >> 5;
  const int hb   = tid * 4;            // 4 contiguous HID columns per thread

  // ---- stage gamma/beta into LDS (CDNA5 async global->LDS path) ----
#if MOLT_ASYNC_LDS
  {
    __builtin_amdgcn_global_load_async_to_lds_b128(
        (molt_gv4i*)(ln_gamma + hb), (molt_lv4i*)(s_gamma + hb), 0, 0);
    __builtin_amdgcn_global_load_async_to_lds_b128(
        (molt_gv4i*)(ln_beta + hb), (molt_lv4i*)(s_beta + hb), 0, 0);
    __builtin_amdgcn_s_wait_asynccnt(0);
  }
#else
  *(float4*)(s_gamma + hb) = *(const float4*)(ln_gamma + hb);
  *(float4*)(s_beta  + hb) = *(const float4*)(ln_beta  + hb);
#endif
  __syncthreads();

  const float4 gv = *(const float4*)(s_gamma + hb);
  const float4 bv = *(const float4*)(s_beta  + hb);

  const bool is_tok = (hb < D);        // threads 0..191: token path
  const int  j  = hb - D;              // threads 192..255: position path
  const int  kk = (j >> 7) & 1;        // which of the K=2 gathered rows
  const int  pp = j & (P - 1);         // column inside that row (16B aligned)

  const int tok0 = blockIdx.x * TOKS;
  const int b    = tok0 >> 9;          // L==512; all TOKS tokens share batch b
  const float* __restrict__ lp_b = lp_embeds + (size_t)b * L * P;

  for (int t = 0; t < TOKS; ++t) {
    const int tok = tok0 + t;          // uniform -> metadata loads become s_load
    float v0, v1, v2, v3;

    if (is_tok) {
      const int iid = input_ids[tok];
      const int tt  = token_type_ids[tok];
      float scale = 1.0f;
      if (tt == 5) scale += mol_features[tok];   // FEAT
      if (tt == 6) scale += target_values[tok];  // TGT (mutually exclusive)

      float4 e = *(const float4*)(emb_table + iid * D + hb);
      v0 = e.x * scale; v1 = e.y * scale; v2 = e.z * scale; v3 = e.w * scale;

#define MOLT_ACC4(base_ptr, row)                                        \
      { float4 _a = *(const float4*)((base_ptr) + (row) * D + hb);      \
        v0 += _a.x; v1 += _a.y; v2 += _a.z; v3 += _a.w; }

      MOLT_ACC4(type_table, tt);
      MOLT_ACC4(ring_tab,   p_ring[tok]);
      MOLT_ACC4(charge_tab, p_charge[tok]);
      MOLT_ACC4(hyb_tab,    p_hyb[tok]);
      MOLT_ACC4(chir_tab,   p_chir[tok]);
      MOLT_ACC4(arom_tab,   p_arom[tok]);
      MOLT_ACC4(conj_tab,   p_conj[tok]);
      MOLT_ACC4(stereo_tab, p_stereo[tok]);
#undef MOLT_ACC4

      if (t + 1 < TOKS) {   // hide next gather's latency (global_prefetch_b8)
        __builtin_prefetch(emb_table + input_ids[tok + 1] * D + hb, 0, 3);
      }
    } else {
      const int pid = pos_embed_ids[tok * K + kk];
      float4 q = *(const float4*)(lp_b + pid * P + pp);
      v0 = (q.x != q.x) ? 0.f : q.x;   // nan_to_num
      v1 = (q.y != q.y) ? 0.f : q.y;
      v2 = (q.z != q.z) ? 0.f : q.z;
      v3 = (q.w != q.w) ? 0.f : q.w;
      if (t + 1 < TOKS) {
        __builtin_prefetch(lp_b + pos_embed_ids[(tok + 1) * K + kk] * P + pp, 0, 3);
      }
    }

    // ---- two-pass LayerNorm over HID=1024 (matches reference structure) ----
    const float mu = block_sum(v0 + v1 + v2 + v3, s_red, lane, wid) * (1.0f / HID);
    const float d0 = v0 - mu, d1 = v1 - mu, d2 = v2 - mu, d3 = v3 - mu;
    const float var =
        block_sum(d0 * d0 + d1 * d1 + d2 * d2 + d3 * d3, s_red, lane, wid) * (1.0f / HID);

    const float a = var + EPS;
    float rs = rsqrtf(a);
    rs = rs * (1.5f - 0.5f * a * rs * rs);   // one Newton step for accuracy

    float4 o;
    o.x = d0 * rs * gv.x + bv.x;
    o.y = d1 * rs * gv.y + bv.y;
    o.z = d2 * rs * gv.z + bv.z;
    o.w = d3 * rs * gv.w + bv.w;
    *(float4*)(out + (size_t)tok * HID + hb) = o;   // coalesced b128 store
  }
}

extern "C" void kernel_launch(void* const* d_in, const int* in_sizes, int n_in,
                              void* d_out, int out_size, void* d_ws, size_t ws_size,
                              hipStream_t stream) {
  (void)in_sizes; (void)n_in; (void)out_size; (void)d_ws; (void)ws_size;

  dim3 grid(NTOK / TOKS);
  dim3 block(TPB);
  molt_embed_ln_kernel<<<grid, block, 0, stream>>>(
      (const int*)  d_in[0],  // input_ids
      (const int*)  d_in[1],  // token_type_ids
      (const int*)  d_in[2],  // pos_embed_ids
      (const float*)d_in[3],  // lp_embeds
      (const int*)  d_in[4],  // prop_atom_in_ring
      (const int*)  d_in[5],  // prop_atom_charge
      (const int*)  d_in[6],  // prop_atom_hybridization
      (const int*)  d_in[7],  // prop_atom_chirality
      (const int*)  d_in[8],  // prop_bond_aromatic
      (const int*)  d_in[9],  // prop_bond_conjugated
      (const int*)  d_in[10], // prop_bond_stereo
      (const float*)d_in[11], // mol_features
      (const float*)d_in[12], // target_values
      (const float*)d_in[13], // emb_table
      (const float*)d_in[14], // type_table
      (const float*)d_in[15], // ring_tab
      (const float*)d_in[16], // charge_tab
      (const float*)d_in[17], // hyb_tab
      (const float*)d_in[18], // chir_tab
      (const float*)d_in[19], // arom_tab
      (const float*)d_in[20], // conj_tab
      (const float*)d_in[21], // stereo_tab
      (const float*)d_in[22], // ln_gamma
      (const float*)d_in[23], // ln_beta
      (float*)d_out);
}